// SimCLRLoss_68358699483154
// MI455X (gfx1250) — compile-verified
//
#include <hip/hip_runtime.h>
#include <math.h>

typedef __attribute__((ext_vector_type(2))) float v2f;
typedef __attribute__((ext_vector_type(8))) float v8f;

#define DIMK 512
#define TEMP_INV 2.0f      // 1 / 0.5
#define NEG_BIG (-1.0e30f) // finite -inf sentinel (logits are in [-4, 4])
#define LDS_PITCH 516      // 512 + 4 pad -> conflict-free ds_load_b64
#define KSTEPS (DIMK / 4)  // 128 wmma_f32_16x16x4 steps per 16x16 tile

// ---------------------------------------------------------------------------
// Kernel 1: row L2-normalize. One wave (32 lanes) per row, float4 loads.
// ---------------------------------------------------------------------------
__global__ __launch_bounds__(256)
void simclr_normalize_kernel(const float* __restrict__ z,
                             float* __restrict__ zn, int N) {
  const int wave = threadIdx.x >> 5;
  const int lane = threadIdx.x & 31;
  const int row  = blockIdx.x * 8 + wave;
  if (row >= N) return;

  const float4* src = (const float4*)(z  + (size_t)row * DIMK);
  float4*       dst = (float4*)(zn + (size_t)row * DIMK);

  float4 vals[4];
  float ss = 0.f;
#pragma unroll
  for (int i = 0; i < 4; ++i) {
    float4 v = src[lane + i * 32];
    vals[i] = v;
    ss += v.x * v.x + v.y * v.y + v.z * v.z + v.w * v.w;
  }
#pragma unroll
  for (int off = 16; off >= 1; off >>= 1)
    ss += __shfl_xor(ss, off, 32);

  const float inv = 1.0f / fmaxf(sqrtf(ss), 1e-12f);
#pragma unroll
  for (int i = 0; i < 4; ++i) {
    float4 v = vals[i];
    v.x *= inv; v.y *= inv; v.z *= inv; v.w *= inv;
    dst[lane + i * 32] = v;
  }
}

// ---------------------------------------------------------------------------
// Kernel 2: fused WMMA GEMM (sim = zn @ zn^T / T) + online masked logsumexp.
// Block = 16 rows. A strip: global -> LDS (once per block) -> VGPRs (once per
// wave, 128 v2f held live across all 32 column tiles). Per tile only the B
// stream from L2 remains: 128 b64 loads feeding 128 v_wmma_f32_16x16x4_f32,
// split into two independent accumulator chains for XDL ILP at low occupancy.
// ---------------------------------------------------------------------------
__global__ __launch_bounds__(256)
void simclr_wmma_lse_kernel(const float* __restrict__ zn,
                            float* __restrict__ rowTerm, int N) {
  const int lane = threadIdx.x & 31;
  const int wave = threadIdx.x >> 5;
  const int l16  = lane & 15;
  const int half = lane >> 4;
  const int rowBase   = blockIdx.x * 16;
  const int nColTiles = N >> 4;
  const int bs        = N >> 1;

  __shared__ float sA[16][LDS_PITCH];   // padded A strip: 16 x 512
  __shared__ float sM[8][16];
  __shared__ float sS[8][16];
  __shared__ float sPos[16];

  // --- cooperative fill of the A strip (16 rows x 128 float4) ---
  for (int idx = threadIdx.x; idx < 16 * 128; idx += 256) {
    const int r  = idx >> 7;         // row 0..15
    const int c4 = idx & 127;        // float4 column
    float4 v = *(const float4*)(zn + (size_t)(rowBase + r) * DIMK + c4 * 4);
    *(float4*)&sA[r][c4 * 4] = v;
  }
  __syncthreads();

  const int kOff = half * 2;         // per-lane K offset for A/B operands

  // --- A strip into registers, once per wave, reused for all 32 tiles ---
  v2f aReg[KSTEPS];
#pragma unroll
  for (int i = 0; i < KSTEPS; ++i)
    aReg[i] = *(const v2f*)&sA[l16][kOff + i * 4];   // ds_load_b64

  float m[8], s[8];
#pragma unroll
  for (int v = 0; v < 8; ++v) { m[v] = NEG_BIG; s[v] = 0.f; }
  float posVal = 0.f;
  int   posRow = -1;

  for (int ct = wave; ct < nColTiles; ct += 8) {
    const int colBase = ct << 4;
    const float* bPtr = zn + (size_t)(colBase + l16) * DIMK + kOff;

    v8f acc0 = {};
    v8f acc1 = {};
#pragma unroll
    for (int i = 0; i < KSTEPS; i += 2) {
      v2f b0 = *(const v2f*)(bPtr + i * 4);
      v2f b1 = *(const v2f*)(bPtr + i * 4 + 4);
      acc0 = __builtin_amdgcn_wmma_f32_16x16x4_f32(
          false, aReg[i],     false, b0, (short)0, acc0, false, false);
      acc1 = __builtin_amdgcn_wmma_f32_16x16x4_f32(
          false, aReg[i + 1], false, b1, (short)0, acc1, false, false);
    }

    const int col = colBase + l16;
#pragma unroll
    for (int v = 0; v < 8; ++v) {
      const int r   = v + half * 8;       // local row in tile
      const int row = rowBase + r;
      float x = (acc0[v] + acc1[v]) * TEMP_INV;
      int pc = row + bs; if (pc >= N) pc -= N;    // positive column
      const bool isPos = (col == pc);
      posVal = isPos ? x : posVal;
      posRow = isPos ? r : posRow;
      x = (col == row) ? NEG_BIG : x;             // mask self-similarity
      // branchless online logsumexp (exp(-huge) == 0 cleans sentinel terms)
      const float nm = fmaxf(m[v], x);
      s[v] = s[v] * __expf(m[v] - nm) + __expf(x - nm);
      m[v] = nm;
    }
  }

  // merge (m, s) across the 16 lanes of each half-wave (same row set)
#pragma unroll
  for (int off = 1; off <= 8; off <<= 1) {
#pragma unroll
    for (int v = 0; v < 8; ++v) {
      float m2 = __shfl_xor(m[v], off, 32);
      float s2 = __shfl_xor(s[v], off, 32);
      float M  = fmaxf(m[v], m2);
      s[v] = s[v] * __expf(m[v] - M) + s2 * __expf(m2 - M);
      m[v] = M;
    }
  }

  if (l16 == 0) {
#pragma unroll
    for (int v = 0; v < 8; ++v) {
      sM[wave][v + half * 8] = m[v];
      sS[wave][v + half * 8] = s[v];
    }
  }
  if (posRow >= 0) sPos[posRow] = posVal;  // exactly one writer per row/block
  __syncthreads();

  if (threadIdx.x < 16) {
    const int r = threadIdx.x;
    float mm = sM[0][r], ss = sS[0][r];
#pragma unroll
    for (int w = 1; w < 8; ++w) {
      float m2 = sM[w][r], s2 = sS[w][r];
      float M  = fmaxf(mm, m2);
      ss = ss * __expf(mm - M) + s2 * __expf(m2 - M);
      mm = M;
    }
    rowTerm[rowBase + r] = (mm + __logf(ss)) - sPos[r];
  }
}

// ---------------------------------------------------------------------------
// Kernel 3: sum N row terms -> scalar loss = sum / N. Deterministic, no atomics.
// ---------------------------------------------------------------------------
__global__ __launch_bounds__(256)
void simclr_reduce_kernel(const float* __restrict__ rowTerm,
                          float* __restrict__ out, int N) {
  __shared__ float sdata[256];
  float sum = 0.f;
  for (int i = threadIdx.x; i < N; i += 256) sum += rowTerm[i];
  sdata[threadIdx.x] = sum;
  __syncthreads();
#pragma unroll
  for (int stride = 128; stride >= 1; stride >>= 1) {
    if ((int)threadIdx.x < stride) sdata[threadIdx.x] += sdata[threadIdx.x + stride];
    __syncthreads();
  }
  if (threadIdx.x == 0) out[0] = sdata[0] / (float)N;
}

// ---------------------------------------------------------------------------
extern "C" void kernel_launch(void* const* d_in, const int* in_sizes, int n_in,
                              void* d_out, int out_size, void* d_ws, size_t ws_size,
                              hipStream_t stream) {
  const float* z = (const float*)d_in[0];
  const int N = in_sizes[0] / DIMK;          // 4096

  float* zn      = (float*)d_ws;             // N*DIMK floats (8 MB)
  float* rowTerm = zn + (size_t)N * DIMK;    // N floats
  float* out     = (float*)d_out;

  simclr_normalize_kernel<<<(N + 7) / 8, 256, 0, stream>>>(z, zn, N);
  simclr_wmma_lse_kernel<<<N / 16, 256, 0, stream>>>(zn, rowTerm, N);
  simclr_reduce_kernel<<<1, 256, 0, stream>>>(rowTerm, out, N);
}